// GAT_54348516164015
// MI455X (gfx1250) — compile-verified
//
#include <hip/hip_runtime.h>
#include <math.h>

typedef __attribute__((ext_vector_type(2))) float v2f;
typedef __attribute__((ext_vector_type(8))) float v8f;

#define GAT_NEG_SLOPE 0.2f
#define GAT_SM_EPS 1e-16f
#define GAT_LN_EPS 1e-5f

// ---------------- wave32 reduction helpers ----------------
__device__ __forceinline__ float warp_sum32(float v) {
#pragma unroll
  for (int o = 16; o > 0; o >>= 1) v += __shfl_xor(v, o, 32);
  return v;
}
__device__ __forceinline__ float warp_max32(float v) {
#pragma unroll
  for (int o = 16; o > 0; o >>= 1) v = fmaxf(v, __shfl_xor(v, o, 32));
  return v;
}

// float atomic max via monotonic int/uint punning (works for all signs)
__device__ __forceinline__ void atomic_max_f32(float* addr, float v) {
  if (v >= 0.0f) atomicMax((int*)addr, __float_as_int(v));
  else           atomicMin((unsigned int*)addr, __float_as_uint(v));
}

// ---------------- generic fill ----------------
__global__ void fill_f32(float* __restrict__ p, int n, float v) {
  int i = blockIdx.x * blockDim.x + threadIdx.x;
  int stride = gridDim.x * blockDim.x;
  for (; i < n; i += stride) p[i] = v;
}

// ---------------- fp32 WMMA GEMM: C[M,Nc] = A[M,K] x B[K,Nc] ----------------
// grid.x = M/16 row-tiles; blockDim.x = 2*Nc (one wave per 16-col tile).
// A tile staged through LDS (padded stride -> conflict-free ds_read).
__global__ void gemm_wmma_f32(const float* __restrict__ A, const float* __restrict__ B,
                              float* __restrict__ C, int M, int K, int Nc) {
  extern __shared__ float As[];           // 16 x (K+1)
  const int KP = K + 1;
  const int mBase = blockIdx.x * 16;
  const int tid = threadIdx.x;

  for (int i = tid; i < 16 * K; i += blockDim.x) {
    int r = i / K, c = i - r * K;
    int row = mBase + r;
    As[r * KP + c] = (row < M) ? A[(size_t)row * K + c] : 0.0f;
  }
  __syncthreads();

  const int wave = tid >> 5;
  const int lane = tid & 31;
  const int nBase = wave * 16;
  const int l15 = lane & 15;
  const int khalf = (lane >> 4) * 2;      // 0 for lanes 0-15, 2 for lanes 16-31

  v8f acc = {};
  for (int k0 = 0; k0 < K; k0 += 4) {
    v2f a, b;
    // A 16x4 frag: lane m=l15, VGPR0->k0+khalf, VGPR1->k0+khalf+1
    a.x = As[l15 * KP + k0 + khalf];
    a.y = As[l15 * KP + k0 + khalf + 1];
    // B 4x16 frag (mirror layout): lane n=l15
    b.x = B[(size_t)(k0 + khalf) * Nc + nBase + l15];
    b.y = B[(size_t)(k0 + khalf + 1) * Nc + nBase + l15];
    acc = __builtin_amdgcn_wmma_f32_16x16x4_f32(false, a, false, b,
                                                (short)0, acc, false, false);
  }

  // C/D 16x16 f32 layout: VGPR r -> row (r + 8*(lane>=16)), col = lane&15
  const int rowTop = mBase + (lane >> 4) * 8;
  const int col = nBase + l15;
#pragma unroll
  for (int r = 0; r < 8; ++r) {
    int row = rowTop + r;
    if (row < M) C[(size_t)row * Nc + col] = acc[r];
  }
}

// ---------------- attention scores: e_src/e_dst [N,H] ----------------
__global__ void attn_scores_k(const float* __restrict__ h, const float* __restrict__ a_src,
                              const float* __restrict__ a_dst, float* __restrict__ es,
                              float* __restrict__ ed, int N, int H, int CH) {
  int idx = blockIdx.x * blockDim.x + threadIdx.x;
  if (idx >= N * H) return;
  int n = idx / H, hh = idx - n * H;
  const float* hp = h + (size_t)n * H * CH + hh * CH;
  const float* asp = a_src + hh * CH;
  const float* adp = a_dst + hh * CH;
  float s = 0.0f, d = 0.0f;
  for (int c = 0; c < CH; ++c) { float t = hp[c]; s += t * asp[c]; d += t * adp[c]; }
  es[idx] = s; ed[idx] = d;
}

__device__ __forceinline__ void edge_sd(const int* __restrict__ ei, int E, int e,
                                        int& s, int& d) {
  if (e < E) { s = ei[e]; d = ei[E + e]; }
  else       { s = d = e - E; }          // self-loop
}

// ---------------- pass A: per-dst segment max ----------------
__global__ void edge_max_k(const int* __restrict__ ei, int E, int N,
                           const float* __restrict__ es, const float* __restrict__ ed,
                           float* __restrict__ m, int H) {
  int idx = blockIdx.x * blockDim.x + threadIdx.x;
  int tot = (E + N) * H;
  if (idx >= tot) return;
  int h = idx & (H - 1);
  int e = idx >> 2;                      // H == 4
  int s, d; edge_sd(ei, E, e, s, d);
  float v = es[s * H + h] + ed[d * H + h];
  v = (v > 0.0f) ? v : GAT_NEG_SLOPE * v;
  atomic_max_f32(&m[d * H + h], v);
}

// ---------------- pass B: per-dst sum of exp ----------------
__global__ void edge_expsum_k(const int* __restrict__ ei, int E, int N,
                              const float* __restrict__ es, const float* __restrict__ ed,
                              const float* __restrict__ m, float* __restrict__ denom, int H) {
  int idx = blockIdx.x * blockDim.x + threadIdx.x;
  int tot = (E + N) * H;
  if (idx >= tot) return;
  int h = idx & (H - 1);
  int e = idx >> 2;
  int s, d; edge_sd(ei, E, e, s, d);
  float v = es[s * H + h] + ed[d * H + h];
  v = (v > 0.0f) ? v : GAT_NEG_SLOPE * v;
  atomicAdd(&denom[d * H + h], __expf(v - m[d * H + h]));
}

// ---------------- pass C: wave-per-edge scatter aggregation ----------------
template <int CHT, int CH>
__global__ void edge_agg_k(const int* __restrict__ ei, int E, int N,
                           const float* __restrict__ feat, const float* __restrict__ es,
                           const float* __restrict__ ed, const float* __restrict__ m,
                           const float* __restrict__ denom, float* __restrict__ out) {
  const int H = CHT / CH;
  int gw = (blockIdx.x * blockDim.x + threadIdx.x) >> 5;
  int lane = threadIdx.x & 31;
  if (gw >= E + N) return;
  int s, d; edge_sd(ei, E, gw, s, d);

  float alpha = 0.0f;
  if (lane < H) {
    float v = es[s * H + lane] + ed[d * H + lane];
    v = (v > 0.0f) ? v : GAT_NEG_SLOPE * v;
    alpha = __expf(v - m[d * H + lane]) / (denom[d * H + lane] + GAT_SM_EPS);
  }
  const float* fp = feat + (size_t)s * CHT;
  float* op = out + (size_t)d * CHT;
#pragma unroll
  for (int c = lane; c < CHT; c += 32) {
    int head = c / CH;
    float a = __shfl(alpha, head, 32);
    atomicAdd(&op[c], fp[c] * a);
  }
}

// ---------------- ELU + LayerNorm over 128 features (wave per node) ----------------
__global__ void elu_ln128_k(float* __restrict__ x, const float* __restrict__ bias,
                            const float* __restrict__ g, const float* __restrict__ b, int N) {
  int node = blockIdx.x * (blockDim.x >> 5) + (threadIdx.x >> 5);
  int lane = threadIdx.x & 31;
  if (node >= N) return;
  float* xp = x + (size_t)node * 128;
  float v[4];
  float sum = 0.0f;
#pragma unroll
  for (int j = 0; j < 4; ++j) {
    int c = lane * 4 + j;
    float t = xp[c] + bias[c];
    t = (t > 0.0f) ? t : (__expf(t) - 1.0f);   // ELU(alpha=1)
    v[j] = t; sum += t;
  }
  float mu = warp_sum32(sum) * (1.0f / 128.0f);
  float vs = 0.0f;
#pragma unroll
  for (int j = 0; j < 4; ++j) { float dd = v[j] - mu; vs += dd * dd; }
  float var = warp_sum32(vs) * (1.0f / 128.0f);
  float rstd = rsqrtf(var + GAT_LN_EPS);
#pragma unroll
  for (int j = 0; j < 4; ++j) {
    int c = lane * 4 + j;
    xp[c] = (v[j] - mu) * rstd * g[c] + b[c];
  }
}

// ---------------- head-mean + bias + LayerNorm + log_softmax (40 cls, wave/node) ----
__global__ void final_k(const float* __restrict__ out2, const float* __restrict__ b2,
                        const float* __restrict__ g, const float* __restrict__ bb,
                        float* __restrict__ y, int N) {
  int node = blockIdx.x * (blockDim.x >> 5) + (threadIdx.x >> 5);
  int lane = threadIdx.x & 31;
  if (node >= N) return;
  const bool has1 = (lane < 8);
  const float* op = out2 + (size_t)node * 160;

  float x0, x1 = 0.0f;
  { float s = 0.0f;
#pragma unroll
    for (int h = 0; h < 4; ++h) s += op[h * 40 + lane];
    x0 = s * 0.25f + b2[lane]; }
  if (has1) {
    float s = 0.0f;
#pragma unroll
    for (int h = 0; h < 4; ++h) s += op[h * 40 + lane + 32];
    x1 = s * 0.25f + b2[lane + 32];
  }
  float mu = warp_sum32(x0 + (has1 ? x1 : 0.0f)) * (1.0f / 40.0f);
  float d0 = x0 - mu, d1 = has1 ? (x1 - mu) : 0.0f;
  float var = warp_sum32(d0 * d0 + d1 * d1) * (1.0f / 40.0f);
  float rstd = rsqrtf(var + GAT_LN_EPS);
  float z0 = d0 * rstd * g[lane] + bb[lane];
  float z1 = has1 ? (d1 * rstd * g[lane + 32] + bb[lane + 32]) : -3.0e38f;
  float mx = warp_max32(fmaxf(z0, z1));
  float se = warp_sum32(__expf(z0 - mx) + (has1 ? __expf(z1 - mx) : 0.0f));
  float lse = __logf(se);
  float* yp = y + (size_t)node * 40;
  yp[lane] = z0 - mx - lse;
  if (has1) yp[lane + 32] = z1 - mx - lse;
}

// =======================================================================
extern "C" void kernel_launch(void* const* d_in, const int* in_sizes, int n_in,
                              void* d_out, int out_size, void* d_ws, size_t ws_size,
                              hipStream_t stream) {
  const float* x        = (const float*)d_in[0];
  const int*   ei       = (const int*)d_in[1];
  const float* W1       = (const float*)d_in[2];
  const float* att_src1 = (const float*)d_in[3];
  const float* att_dst1 = (const float*)d_in[4];
  const float* b1       = (const float*)d_in[5];
  const float* W2       = (const float*)d_in[6];
  const float* att_src2 = (const float*)d_in[7];
  const float* att_dst2 = (const float*)d_in[8];
  const float* b2       = (const float*)d_in[9];
  const float* ln0_g    = (const float*)d_in[10];
  const float* ln0_b    = (const float*)d_in[11];
  const float* ln1_g    = (const float*)d_in[12];
  const float* ln1_b    = (const float*)d_in[13];

  const int F_IN = 256, H1C = 128, H2C = 160, H = 4;
  const int N = in_sizes[0] / F_IN;      // 50000
  const int E = in_sizes[1] / 2;         // 800000
  const int EL = E + N;                  // edges incl. self-loops

  float* ws   = (float*)d_ws;
  float* h1   = ws;                              // N*128
  float* out1 = h1   + (size_t)N * H1C;          // N*128 (post-agg, then post-LN in place)
  float* out2 = out1 + (size_t)N * H1C;          // N*160
  float* h2   = out2 + (size_t)N * H2C;          // N*160
  float* es   = h2   + (size_t)N * H2C;          // N*4
  float* ed   = es   + (size_t)N * H;            // N*4
  float* mseg = ed   + (size_t)N * H;            // N*4
  float* dseg = mseg + (size_t)N * H;            // N*4

  const float NEG_INF = -__builtin_inff();
  const int mTiles = (N + 15) / 16;

  // ----- conv1 -----
  gemm_wmma_f32<<<mTiles, 2 * H1C, 16 * (F_IN + 1) * sizeof(float), stream>>>(
      x, W1, h1, N, F_IN, H1C);
  attn_scores_k<<<(N * H + 255) / 256, 256, 0, stream>>>(h1, att_src1, att_dst1,
                                                         es, ed, N, H, 32);
  fill_f32<<<512, 256, 0, stream>>>(mseg, N * H, NEG_INF);
  fill_f32<<<512, 256, 0, stream>>>(dseg, N * H, 0.0f);
  fill_f32<<<2048, 256, 0, stream>>>(out1, N * H1C, 0.0f);
  edge_max_k<<<(EL * H + 255) / 256, 256, 0, stream>>>(ei, E, N, es, ed, mseg, H);
  edge_expsum_k<<<(EL * H + 255) / 256, 256, 0, stream>>>(ei, E, N, es, ed, mseg, dseg, H);
  edge_agg_k<128, 32><<<(EL + 7) / 8, 256, 0, stream>>>(ei, E, N, h1, es, ed,
                                                        mseg, dseg, out1);
  elu_ln128_k<<<(N + 7) / 8, 256, 0, stream>>>(out1, b1, ln0_g, ln0_b, N);

  // ----- conv2 -----
  gemm_wmma_f32<<<mTiles, 2 * H2C, 16 * (H1C + 1) * sizeof(float), stream>>>(
      out1, W2, h2, N, H1C, H2C);
  attn_scores_k<<<(N * H + 255) / 256, 256, 0, stream>>>(h2, att_src2, att_dst2,
                                                         es, ed, N, H, 40);
  fill_f32<<<512, 256, 0, stream>>>(mseg, N * H, NEG_INF);
  fill_f32<<<512, 256, 0, stream>>>(dseg, N * H, 0.0f);
  fill_f32<<<2048, 256, 0, stream>>>(out2, N * H2C, 0.0f);
  edge_max_k<<<(EL * H + 255) / 256, 256, 0, stream>>>(ei, E, N, es, ed, mseg, H);
  edge_expsum_k<<<(EL * H + 255) / 256, 256, 0, stream>>>(ei, E, N, es, ed, mseg, dseg, H);
  edge_agg_k<160, 40><<<(EL + 7) / 8, 256, 0, stream>>>(ei, E, N, h2, es, ed,
                                                        mseg, dseg, out2);

  // ----- head mean + bias + LN + log_softmax -----
  final_k<<<(N + 7) / 8, 256, 0, stream>>>(out2, b2, ln1_g, ln1_b, (float*)d_out, N);
}